// GHMC_57123065037106
// MI455X (gfx1250) — compile-verified
//
#include <hip/hip_runtime.h>
#include <hip/hip_bf16.h>

// GHM-C loss, fused single-pass:
//   loss = -(1/n) * sum_b S_b / cnt_b   (n = #non-empty bins)
// Memory-bound: 536 MB streamed once -> ~23 us floor at 23.3 TB/s.

#define BINS      10
#define THREADS   256
#define HSTRIDE   32          // dwords per per-lane histogram (10 sums @0, 10 counts @16, pad)
#define NBLOCKS   2048

typedef float f4  __attribute__((ext_vector_type(4)));
typedef float v2f __attribute__((ext_vector_type(2)));
typedef float v8f __attribute__((ext_vector_type(8)));

__global__ void ghm_zero_ws(float* gsum, unsigned* gcnt) {
    int t = threadIdx.x;
    if (t < BINS) { gsum[t] = 0.0f; gcnt[t] = 0u; }
}

__device__ __forceinline__ void ghm_elem(float* __restrict__ my, float p, float t) {
    // g = |p - t|
    float g = fabsf(p - t);
    // bce = t*log_sigmoid(p) + (1-t)*log_sigmoid(-p)
    //     = -log1p(exp(-|p|)) + t*min(p,0) + (1-t)*min(-p,0)
    float ap  = fabsf(p);
    float l1p = __logf(1.0f + __expf(-ap));          // v_exp_f32 / v_log_f32 (TRANS)
    float bce = fmaf(t, fminf(p, 0.0f),
                fmaf(1.0f - t, fminf(-p, 0.0f), -l1p));
    // bin: edges[i] <= g < edges[i+1]; last edge = 1 + 1e-6
    if (g < 1.0f + 1e-6f) {
        int b = (int)(g * 10.0f);
        b = b > 9 ? 9 : b;
        my[b]      += bce;    // per-lane private LDS slot: no conflicts, no atomics
        my[16 + b] += 1.0f;   // per-lane count (<= few hundred per block -> f32 exact)
    }
}

__global__ void __launch_bounds__(THREADS)
ghm_pass(const f4* __restrict__ pred, const f4* __restrict__ targ,
         float* __restrict__ gsum, unsigned* __restrict__ gcnt, int n4) {
    __shared__ float hist[THREADS * HSTRIDE];       // 32 KB

    for (int i = threadIdx.x; i < THREADS * HSTRIDE; i += THREADS)
        hist[i] = 0.0f;
    __syncthreads();

    float* my = &hist[threadIdx.x * HSTRIDE];

    int i0     = blockIdx.x * blockDim.x + threadIdx.x;
    int stride = gridDim.x * blockDim.x;
    for (int i = i0; i < n4; i += stride) {
        // streamed once, bigger than L2 -> non-temporal B128 loads
        f4 p = __builtin_nontemporal_load(&pred[i]);
        f4 t = __builtin_nontemporal_load(&targ[i]);
        ghm_elem(my, p.x, t.x);
        ghm_elem(my, p.y, t.y);
        ghm_elem(my, p.z, t.z);
        ghm_elem(my, p.w, t.w);
    }
    __syncthreads();

    // ---- Block reduction via the matrix engine -------------------------------
    // hist is a 256 x 20 matrix (stride 32); we need column sums.
    // V_WMMA_F32_16X16X4_F32 with A = ones(16x4): D[m][n] = sum_k B[k][n] + C[m][n],
    // i.e. an accumulating 4-row column-sum. 64 WMMAs collapse all 256 rows.
    // Wave 0 only; branch is wave-uniform so EXEC is all ones inside (WMMA req).
    if (threadIdx.x < 32) {
        int lane = threadIdx.x;
        int col  = lane & 15;
        int khi  = (lane >> 4) << 1;                // B layout: VGPR0 -> K=0/2, VGPR1 -> K=1/3
        v2f a;  a.x = 1.0f; a.y = 1.0f;             // ones(16x4) fragment

        #pragma unroll
        for (int grp = 0; grp < 2; ++grp) {         // grp 0: sums, grp 1: counts
            int goff = grp * 16;
            v8f acc = {};
            for (int k = 0; k < THREADS; k += 4) {
                v2f b;
                b.x = hist[(k + khi + 0) * HSTRIDE + goff + col];
                b.y = hist[(k + khi + 1) * HSTRIDE + goff + col];
                acc = __builtin_amdgcn_wmma_f32_16x16x4_f32(
                          /*neg_a=*/false, a, /*neg_b=*/false, b,
                          /*c_mod=*/(short)0, acc,
                          /*reuse_a=*/false, /*reuse_b=*/false);
            }
            // C/D layout: VGPR0, lanes 0-15 hold M=0, N=lane -> acc[0] = colsum(N=lane)
            if (lane < BINS) {
                float v = acc[0];
                if (grp == 0) atomicAdd(&gsum[lane], v);
                else          atomicAdd(&gcnt[lane], (unsigned)(v + 0.5f));
            }
        }
    }
}

__global__ void ghm_finalize(const float* __restrict__ gsum,
                             const unsigned* __restrict__ gcnt,
                             float* __restrict__ out) {
    if (threadIdx.x == 0) {
        float total = 0.0f;
        int   n     = 0;
        #pragma unroll
        for (int b = 0; b < BINS; ++b) {
            unsigned c = gcnt[b];
            if (c > 0u) { ++n; total += gsum[b] / (float)c; }
        }
        float nn = (n > 0) ? (float)n : 1.0f;
        out[0] = -total / nn;
    }
}

extern "C" void kernel_launch(void* const* d_in, const int* in_sizes, int n_in,
                              void* d_out, int out_size, void* d_ws, size_t ws_size,
                              hipStream_t stream) {
    const f4* pred = (const f4*)d_in[0];
    const f4* targ = (const f4*)d_in[1];
    float*    gsum = (float*)d_ws;
    unsigned* gcnt = (unsigned*)((float*)d_ws + 16);
    int n  = in_sizes[0];            // 16384*4096 = 67,108,864 (divisible by 4)
    int n4 = n >> 2;

    ghm_zero_ws<<<1, 64, 0, stream>>>(gsum, gcnt);
    ghm_pass<<<NBLOCKS, THREADS, 0, stream>>>(pred, targ, gsum, gcnt, n4);
    ghm_finalize<<<1, 32, 0, stream>>>(gsum, gcnt, (float*)d_out);
}